// Decoder_1228360647103
// MI455X (gfx1250) — compile-verified
//
#include <hip/hip_runtime.h>
#include <cstdint>

// Problem constants
#define B_    256
#define NV_   35709
#define NF_   70789
#define NK_   68
#define NV3   (NV_*3)              // 107127
#define MT    ((NV3+15)/16)        // 6696 M-tiles (exactly 837*8)

// d_out layout (flat float/int32 elements, tuple order: fp, fc, lm, tri)
#define FP_OFF  0
#define FC_OFF  (B_*NV3)                 // 27,424,512
#define LM_OFF  (2*B_*NV3)               // 54,849,024
#define TRI_OFF (2*B_*NV3 + B_*NK_*3)    // 54,901,248

// ws layout (float elements)
#define WS_CTP    0                       // paired coeff transpose: 112*256*2 = 57344
#define CTP_FLOATS (112*256*2)            // 57344 floats = 229376 bytes
#define WS_CENTER 57344                   // 3 (+pad)
#define WS_ROT    57352                   // B*9
#define WS_G      (WS_ROT + B_*9)         // B*27
#define WS_TR     (WS_G + B_*27)          // B*3

typedef float v2f __attribute__((ext_vector_type(2)));
typedef float v8f __attribute__((ext_vector_type(8)));

__device__ __forceinline__ v8f wmma_f32(v2f a, v2f b, v8f c) {
  // D = A(16x4 f32) * B(4x16 f32) + C ; emits v_wmma_f32_16x16x4_f32
  return __builtin_amdgcn_wmma_f32_16x16x4_f32(false, a, false, b, (short)0, c,
                                               false, false);
}

// ---------------------------------------------------------------------------
// K0: meanshape centroid, coeff transpose (paired for B-operand 64b loads),
//     per-batch rotation^T, SH gamma (transposed, +0.8 on DC), translation.
// ---------------------------------------------------------------------------
__global__ void k0_prep(const float* __restrict__ coeff,
                        const float* __restrict__ meanshape,
                        float* __restrict__ ws) {
  __shared__ float red[256 * 3];
  const int t = threadIdx.x;

  // centroid of meanshape over vertices, per coordinate
  float s0 = 0.f, s1 = 0.f, s2 = 0.f;
  for (int v = t; v < NV_; v += 256) {
    s0 += meanshape[3 * v + 0];
    s1 += meanshape[3 * v + 1];
    s2 += meanshape[3 * v + 2];
  }
  red[t * 3 + 0] = s0; red[t * 3 + 1] = s1; red[t * 3 + 2] = s2;
  __syncthreads();
  if (t == 0) {
    float a = 0.f, b = 0.f, c = 0.f;
    for (int i = 0; i < 256; ++i) { a += red[i*3]; b += red[i*3+1]; c += red[i*3+2]; }
    ws[WS_CENTER + 0] = a / (float)NV_;
    ws[WS_CENTER + 1] = b / (float)NV_;
    ws[WS_CENTER + 2] = c / (float)NV_;
  }

  // paired transpose: ctp[k/2][n][k&1] = coeff[n][k], k in [0,224)
  for (int idx = t; idx < 224 * 256; idx += 256) {
    int k = idx >> 8, n = idx & 255;
    ws[WS_CTP + ((size_t)(k >> 1) * 256 + n) * 2 + (k & 1)] = coeff[n * 257 + k];
  }

  // per-batch small math (one thread per batch; blockDim==256==B_)
  {
    const int b = t;
    const float* cb = coeff + b * 257;
    float ax = cb[224], ay = cb[225], az = cb[226];
    float sx = sinf(ax), cx = cosf(ax);
    float sy = sinf(ay), cy = cosf(ay);
    float sz = sinf(az), cz = cosf(az);
    // M = Rz @ Ry @ Rx
    float M00 =  cz * cy, M01 = cz * sy * sx - sz * cx, M02 = cz * sy * cx + sz * sx;
    float M10 =  sz * cy, M11 = sz * sy * sx + cz * cx, M12 = sz * sy * cx - cz * sx;
    float M20 = -sy,      M21 = cy * sx,                M22 = cy * cx;
    // rotation = M^T ; store rot[i][j] = M[j][i]
    float* R = ws + WS_ROT + b * 9;
    R[0] = M00; R[1] = M10; R[2] = M20;
    R[3] = M01; R[4] = M11; R[5] = M21;
    R[6] = M02; R[7] = M12; R[8] = M22;
    // g: (3,9) += 0.8 on [:,0], then transposed to [k][c]
    float* G = ws + WS_G + b * 27;
    for (int c = 0; c < 3; ++c)
      for (int k = 0; k < 9; ++k)
        G[k * 3 + c] = cb[227 + c * 9 + k] + (k == 0 ? 0.8f : 0.0f);
    float* T = ws + WS_TR + b * 3;
    T[0] = cb[254]; T[1] = cb[255]; T[2] = cb[256];
  }
}

// ---------------------------------------------------------------------------
// K1: fused f32 WMMA GEMM. One wave per 16-row M-tile; A register-resident,
//     looped over 16 batch tiles. B operands staged in LDS (229 KB fits in the
//     320 KB/WGP CDNA5 LDS) so coefficient traffic hits L2 once per block.
//     face_shape -> d_out[FP], face_tex -> d_out[FC].
// ---------------------------------------------------------------------------
__global__ void __launch_bounds__(256) k1_gemm(
    const float* __restrict__ idBase, const float* __restrict__ exBase,
    const float* __restrict__ texBase, const float* __restrict__ meanshape,
    const float* __restrict__ meantex, const float* __restrict__ ws,
    float* __restrict__ out) {
  extern __shared__ float ldsc[];   // CTP_FLOATS floats (229,376 bytes)

  // Cooperative LDS fill: whole ctp block, float4-wide (uniform, no divergence)
  {
    const float4* src = (const float4*)(ws + WS_CTP);
    float4* dst = (float4*)ldsc;
#pragma unroll 4
    for (int i = threadIdx.x; i < CTP_FLOATS / 4; i += 256) dst[i] = src[i];
  }
  __syncthreads();

  const int lane  = threadIdx.x & 31;
  const int wave  = threadIdx.x >> 5;
  const int mtile = blockIdx.x * 8 + wave;

  const int r  = lane & 15;   // A row-in-tile / B & D column (N)
  const int kh = lane >> 4;   // K half: lanes 0-15 hold K pair {0,1}, 16-31 {2,3}

  const int mrow = mtile * 16 + r;
  const int mcl  = mrow < NV3 ? mrow : NV3 - 1;   // clamp last partial tile

  // Preload A operands (register resident, reused by all 16 N-tiles)
  v2f aid[20], aex[16], atx[20];
#pragma unroll
  for (int t2 = 0; t2 < 20; ++t2)
    aid[t2] = *(const v2f*)(idBase + (size_t)mcl * 80 + 4 * t2 + 2 * kh);
#pragma unroll
  for (int t2 = 0; t2 < 16; ++t2)
    aex[t2] = *(const v2f*)(exBase + (size_t)mcl * 64 + 4 * t2 + 2 * kh);
#pragma unroll
  for (int t2 = 0; t2 < 20; ++t2)
    atx[t2] = *(const v2f*)(texBase + (size_t)mcl * 80 + 4 * t2 + 2 * kh);

  // Epilogue constants: D rows for this lane are mD0..mD0+7
  const float c0 = ws[WS_CENTER], c1 = ws[WS_CENTER + 1], c2 = ws[WS_CENTER + 2];
  const int mD0 = mtile * 16 + 8 * kh;
  float msv[8], mtv[8];
#pragma unroll
  for (int i = 0; i < 8; ++i) {
    int m  = mD0 + i;
    int mc = m < NV3 ? m : NV3 - 1;
    int cc = mc % 3;
    float ctr = (cc == 0) ? c0 : ((cc == 1) ? c1 : c2);
    msv[i] = meanshape[mc] - ctr;
    mtv[i] = meantex[mc];
  }

  for (int nt = 0; nt < 16; ++nt) {
    const int n = nt * 16 + r;
    v8f accS = {0.f,0.f,0.f,0.f,0.f,0.f,0.f,0.f};
    v8f accT = {0.f,0.f,0.f,0.f,0.f,0.f,0.f,0.f};
#pragma unroll
    for (int t2 = 0; t2 < 20; ++t2) {   // idBase, K = 0..79
      v2f b = *(const v2f*)(ldsc + ((2 * t2 + kh) * 256 + n) * 2);
      accS = wmma_f32(aid[t2], b, accS);
    }
#pragma unroll
    for (int t2 = 0; t2 < 16; ++t2) {   // exBase, coeff cols 80..143
      v2f b = *(const v2f*)(ldsc + ((40 + 2 * t2 + kh) * 256 + n) * 2);
      accS = wmma_f32(aex[t2], b, accS);
    }
#pragma unroll
    for (int t2 = 0; t2 < 20; ++t2) {   // texBase, coeff cols 144..223
      v2f b = *(const v2f*)(ldsc + ((72 + 2 * t2 + kh) * 256 + n) * 2);
      accT = wmma_f32(atx[t2], b, accT);
    }
    float* fpo = out + FP_OFF + (size_t)n * NV3;
    float* fco = out + FC_OFF + (size_t)n * NV3;
#pragma unroll
    for (int i = 0; i < 8; ++i) {
      int m = mD0 + i;
      if (m < NV3) {
        fpo[m] = accS[i] + msv[i];
        fco[m] = accT[i] + mtv[i];
      }
    }
  }
}

// ---------------------------------------------------------------------------
// K2: vertex normals (recompute adjacent face cross products; face id NF_ is
//     the zero pad), SH lighting, color = clip(tex*light/255) in place in FC.
// ---------------------------------------------------------------------------
__global__ void k2_normcolor(const int* __restrict__ tri,
                             const int* __restrict__ point_buf,
                             const float* __restrict__ ws,
                             float* __restrict__ out) {
  const int v = blockIdx.x * blockDim.x + threadIdx.x;
  const int b = blockIdx.y;
  if (v >= NV_) return;
  const float* fs = out + FP_OFF + (size_t)b * NV3;   // staged face_shape

  float ax = 0.f, ay = 0.f, az = 0.f;
#pragma unroll
  for (int j = 0; j < 8; ++j) {
    int f = point_buf[v * 8 + j];
    if (f < NF_) {
      int i0 = tri[3 * f], i1 = tri[3 * f + 1], i2 = tri[3 * f + 2];
      float x1 = fs[3*i0], y1 = fs[3*i0+1], z1 = fs[3*i0+2];
      float x2 = fs[3*i1], y2 = fs[3*i1+1], z2 = fs[3*i1+2];
      float x3 = fs[3*i2], y3 = fs[3*i2+1], z3 = fs[3*i2+2];
      float e1x = x1 - x2, e1y = y1 - y2, e1z = z1 - z2;
      float e2x = x2 - x3, e2y = y2 - y3, e2z = z2 - z3;
      ax += e1y * e2z - e1z * e2y;
      ay += e1z * e2x - e1x * e2z;
      az += e1x * e2y - e1y * e2x;
    }
  }
  float inv = 1.0f / sqrtf(ax * ax + ay * ay + az * az);
  float nx = ax * inv, ny = ay * inv, nz = az * inv;

  const float KA = 1.7724538509055159f;   // a1*c1 = sqrt(pi)
  const float KB = 2.4270323258f;         // a2*c2
  float Y0 = 0.8862269254527581f;         // a0*c0
  float Y1 = -KA * ny, Y2 = KA * nz, Y3 = -KA * nx;
  float Y4 = KB * nx * ny, Y5 = -KB * ny * nz;
  float Y6 = KB * 0.2886751346f * (3.0f * nz * nz - 1.0f);
  float Y7 = -KB * nx * nz;
  float Y8 = KB * 0.5f * (nx * nx - ny * ny);

  const float* g = ws + WS_G + b * 27;            // [k][c]
  float* fc = out + FC_OFF + (size_t)b * NV3 + 3 * v;
#pragma unroll
  for (int c = 0; c < 3; ++c) {
    float L = Y0 * g[0*3+c] + Y1 * g[1*3+c] + Y2 * g[2*3+c] + Y3 * g[3*3+c] +
              Y4 * g[4*3+c] + Y5 * g[5*3+c] + Y6 * g[6*3+c] + Y7 * g[7*3+c] +
              Y8 * g[8*3+c];
    float val = fc[c] * L * (1.0f / 255.0f);
    val = val < 0.f ? 0.f : (val > 1.f ? 1.f : val);
    fc[c] = val;   // in-place: only this thread touches fc[b][v]
  }
}

// ---------------------------------------------------------------------------
// K3: projection = face_shape @ rotation + t, in place in FP slot.
// ---------------------------------------------------------------------------
__global__ void k3_proj(const float* __restrict__ ws, float* __restrict__ out) {
  const int v = blockIdx.x * blockDim.x + threadIdx.x;
  const int b = blockIdx.y;
  if (v >= NV_) return;
  float* p = out + FP_OFF + (size_t)b * NV3 + 3 * v;
  float x = p[0], y = p[1], z = p[2];
  const float* R = ws + WS_ROT + b * 9;
  const float* T = ws + WS_TR + b * 3;
  p[0] = x * R[0] + y * R[3] + z * R[6] + T[0];
  p[1] = x * R[1] + y * R[4] + z * R[7] + T[1];
  p[2] = x * R[2] + y * R[5] + z * R[8] + T[2];
}

// ---------------------------------------------------------------------------
// K4: landmarks gather from projected shape + tri passthrough (int32 bits).
// ---------------------------------------------------------------------------
__global__ void k4_tail(const int* __restrict__ keypoints,
                        const int* __restrict__ tri,
                        float* __restrict__ out) {
  const int idx = blockIdx.x * blockDim.x + threadIdx.x;
  if (idx < B_ * NK_) {
    int b = idx / NK_, k = idx % NK_;
    int vk = keypoints[k];
    const float* p = out + FP_OFF + (size_t)b * NV3 + 3 * vk;
    float* lm = out + LM_OFF + (size_t)idx * 3;
    lm[0] = p[0]; lm[1] = p[1]; lm[2] = p[2];
  } else {
    int j = idx - B_ * NK_;
    if (j < NF_ * 3) ((int*)out)[TRI_OFF + j] = tri[j];
  }
}

// ---------------------------------------------------------------------------
extern "C" void kernel_launch(void* const* d_in, const int* in_sizes, int n_in,
                              void* d_out, int out_size, void* d_ws, size_t ws_size,
                              hipStream_t stream) {
  const float* coeff     = (const float*)d_in[0];
  const float* meanshape = (const float*)d_in[1];
  const float* idBase    = (const float*)d_in[2];
  const float* exBase    = (const float*)d_in[3];
  const float* meantex   = (const float*)d_in[4];
  const float* texBase   = (const float*)d_in[5];
  const int*   tri       = (const int*)d_in[6];
  const int*   point_buf = (const int*)d_in[7];
  const int*   keypoints = (const int*)d_in[8];
  float* out = (float*)d_out;
  float* ws  = (float*)d_ws;

  k0_prep<<<1, 256, 0, stream>>>(coeff, meanshape, ws);

  // 229,376 bytes of dynamic LDS for the staged coefficient block
  k1_gemm<<<MT / 8, 256, CTP_FLOATS * sizeof(float), stream>>>(
      idBase, exBase, texBase, meanshape, meantex, ws, out);

  dim3 g2((NV_ + 255) / 256, B_);
  k2_normcolor<<<g2, 256, 0, stream>>>(tri, point_buf, ws, out);
  k3_proj<<<g2, 256, 0, stream>>>(ws, out);

  int tot = B_ * NK_ + NF_ * 3;
  k4_tail<<<(tot + 255) / 256, 256, 0, stream>>>(keypoints, tri, out);
}